// layers_66288525247136
// MI455X (gfx1250) — compile-verified
//
#include <hip/hip_runtime.h>

// ---------------------------------------------------------------------------
// graph_inception x2 for MI455X (gfx1250, wave32)
//   - SpMM: wave-per-edge scatter with native global_atomic_add_f32
//   - Dense GEMMs: V_WMMA_F32_16X16X4_F32 (full fp32, matches reference)
//   - A-tiles staged global->LDS with GLOBAL_LOAD_ASYNC_TO_LDS_B128 (ASYNCcnt)
//   - Linear-layer algebra collapsed: out = S1@W1^T + S2@W2^T + HOP*(b1+b2)
// ---------------------------------------------------------------------------

#define NN 40000
#define EE 640000
#define DD 128
#define APAD (DD + 4)   // A-tile row pad: lane stride 132 -> banks 4*l+k, conflict-free

typedef __attribute__((ext_vector_type(2))) float v2f;
typedef __attribute__((ext_vector_type(8))) float v8f;

// ---------------------------- zero fill ------------------------------------
__global__ void zero_f32(float* __restrict__ p, int n4) {
  int i = blockIdx.x * blockDim.x + threadIdx.x;
  if (i < n4) {
    float4 z; z.x = 0.f; z.y = 0.f; z.z = 0.f; z.w = 0.f;
    ((float4*)p)[i] = z;
  }
}

// ---------------------------- SpMM (COO scatter) ---------------------------
// y[idx_out[e], :] += vals[e] * x[idx_in[e], :]
// One wave (32 lanes) per edge; each lane owns 4 of the 128 features.
__global__ void spmm_atomic(const int* __restrict__ idx_out,
                            const int* __restrict__ idx_in,
                            const float* __restrict__ vals,
                            const float* __restrict__ x,
                            float* __restrict__ y) {
  int t = blockIdx.x * blockDim.x + threadIdx.x;
  int e = t >> 5;
  int lane = t & 31;
  if (e >= EE) return;
  int r = idx_out[e];
  int c = idx_in[e];
  float v = vals[e];
  float4 xv = ((const float4*)(x + (size_t)c * DD))[lane];
  float* yp = y + (size_t)r * DD + lane * 4;
  unsafeAtomicAdd(yp + 0, v * xv.x);   // global_atomic_add_f32, no return
  unsafeAtomicAdd(yp + 1, v * xv.y);
  unsafeAtomicAdd(yp + 2, v * xv.z);
  unsafeAtomicAdd(yp + 3, v * xv.w);
}

// ------------------- elementwise hop-0 fusion ------------------------------
// lm = a0*fl; rm = b0*fr; Sl2 = lm; Sr2 = rm; Kl = a0+lm; Kr = b0+rm
__global__ void ew_hop0(const float* __restrict__ a0, const float* __restrict__ b0,
                        const float* __restrict__ fl, const float* __restrict__ fr,
                        float* __restrict__ Sl2, float* __restrict__ Sr2,
                        float* __restrict__ Kl, float* __restrict__ Kr, int n4) {
  int i = blockIdx.x * blockDim.x + threadIdx.x;
  if (i >= n4) return;
  float4 a = ((const float4*)a0)[i];
  float4 b = ((const float4*)b0)[i];
  float4 xl = ((const float4*)fl)[i];
  float4 xr = ((const float4*)fr)[i];
  float4 lm, rm, kl, kr;
  lm.x = a.x * xl.x; lm.y = a.y * xl.y; lm.z = a.z * xl.z; lm.w = a.w * xl.w;
  rm.x = b.x * xr.x; rm.y = b.y * xr.y; rm.z = b.z * xr.z; rm.w = b.w * xr.w;
  kl.x = a.x + lm.x; kl.y = a.y + lm.y; kl.z = a.z + lm.z; kl.w = a.w + lm.w;
  kr.x = b.x + rm.x; kr.y = b.y + rm.y; kr.z = b.z + rm.z; kr.w = b.w + rm.w;
  ((float4*)Sl2)[i] = lm;
  ((float4*)Sr2)[i] = rm;
  ((float4*)Kl)[i]  = kl;
  ((float4*)Kr)[i]  = kr;
}

// ------------------- elementwise hop-1 accumulate --------------------------
// S1 += a1; S2 += a1*K
__global__ void ew_acc(float* __restrict__ S1, float* __restrict__ S2,
                       const float* __restrict__ a1, const float* __restrict__ K,
                       int n4) {
  int i = blockIdx.x * blockDim.x + threadIdx.x;
  if (i >= n4) return;
  float4 s1 = ((const float4*)S1)[i];
  float4 s2 = ((const float4*)S2)[i];
  float4 a  = ((const float4*)a1)[i];
  float4 k  = ((const float4*)K)[i];
  s1.x += a.x; s1.y += a.y; s1.z += a.z; s1.w += a.w;
  s2.x += a.x * k.x; s2.y += a.y * k.y; s2.z += a.z * k.z; s2.w += a.w * k.w;
  ((float4*)S1)[i] = s1;
  ((float4*)S2)[i] = s2;
}

// ------------------- fused dual-GEMM + bias + ReLU -------------------------
// out = relu?( S1 @ W1^T + S2 @ W2^T + 2*(b1+b2) )
// Block: 256 threads = 8 waves. Block owns 16 rows; wave w owns columns
// [16w, 16w+16). W1/W2 transposed into padded LDS once per block; the block's
// 16x128 S1/S2 tiles are staged into LDS ONCE via async-to-LDS loads (instead
// of each of the 8 waves re-loading them from global), so the inner loop is
// pure LDS + WMMA.
// WMMA f32 16x16x4 operand layout (ISA 7.12.2):
//   A: lane<16 -> row M=lane,  K={k,k+1}; lane>=16 -> M=lane-16, K={k+2,k+3}
//   B (mirrored): lane half picks rows {k,k+1} / {k+2,k+3}, N = lane&15
//   C/D: vreg v -> row r0 + (lane>=16 ? 8 : 0) + v, col j0 + (lane&15)
__global__ __launch_bounds__(256)
void wmma_gemm2(const float* __restrict__ S1, const float* __restrict__ S2,
                const float* __restrict__ W1, const float* __restrict__ W2,
                const float* __restrict__ b1, const float* __restrict__ b2,
                float* __restrict__ out, int relu_flag) {
  __shared__ float Wt1[DD][DD + 1];   // +1 pad: transpose writes hit distinct banks
  __shared__ float Wt2[DD][DD + 1];
  __shared__ float At1[16][APAD];     // block's S1 tile (16 rows x 128)
  __shared__ float At2[16][APAD];     // block's S2 tile

  const int r0 = blockIdx.x * 16;
  const int t  = threadIdx.x;

  // ---- async-stage the two 16x128 A tiles straight into LDS (ASYNCcnt) ----
  // 512 float4 per tile; 256 threads -> 2 per thread per tile. Coalesced.
  for (int f = t; f < 512; f += 256) {
    int row = f >> 5;          // 0..15
    int c4  = (f & 31) << 2;   // 0,4,...,124
    const float* g1 = S1 + (size_t)(r0 + row) * DD + c4;
    const float* g2 = S2 + (size_t)(r0 + row) * DD + c4;
    unsigned l1 = (unsigned)(uintptr_t)&At1[row][c4];
    unsigned l2 = (unsigned)(uintptr_t)&At2[row][c4];
    asm volatile("global_load_async_to_lds_b128 %0, %1, off"
                 :: "v"(l1), "v"(g1) : "memory");
    asm volatile("global_load_async_to_lds_b128 %0, %1, off"
                 :: "v"(l2), "v"(g2) : "memory");
  }

  // ---- cooperative transposed weight load (VGPR-mediated, needs transpose) ----
  for (int idx = t; idx < DD * DD; idx += 256) {
    int j = idx >> 7;       // output-neuron row of W
    int k = idx & (DD - 1); // input feature
    Wt1[k][j] = W1[idx];    // Wt[k][j] = W[j][k]  (== B operand, row-major)
    Wt2[k][j] = W2[idx];
  }

  asm volatile("s_wait_asynccnt 0x0" ::: "memory");  // our async LDS writes done
  __syncthreads();                                   // everyone's writes visible

  const int lane = t & 31;
  const int wave = t >> 5;
  const int half = lane >> 4;    // 0: K lo pair, 1: K hi pair
  const int l16  = lane & 15;
  const int j0 = wave * 16;

  v8f acc = {};
#pragma unroll
  for (int k = 0; k < DD; k += 4) {
    int ka = k + half * 2;
    v2f a = *(const v2f*)(&At1[l16][ka]);           // A[M=l16, ka..ka+1] from LDS
    v2f b;
    b.x = Wt1[ka][j0 + l16];                        // B[ka,   j0+l16]
    b.y = Wt1[ka + 1][j0 + l16];                    // B[ka+1, j0+l16]
    acc = __builtin_amdgcn_wmma_f32_16x16x4_f32(false, a, false, b,
                                                (short)0, acc, false, false);
  }
#pragma unroll
  for (int k = 0; k < DD; k += 4) {
    int ka = k + half * 2;
    v2f a = *(const v2f*)(&At2[l16][ka]);
    v2f b;
    b.x = Wt2[ka][j0 + l16];
    b.y = Wt2[ka + 1][j0 + l16];
    acc = __builtin_amdgcn_wmma_f32_16x16x4_f32(false, a, false, b,
                                                (short)0, acc, false, false);
  }

  const float bias = 2.0f * (b1[j0 + l16] + b2[j0 + l16]);   // HOP copies of each bias
  float* op = out + (size_t)(r0 + half * 8) * DD + j0 + l16;
#pragma unroll
  for (int v = 0; v < 8; v++) {
    float val = acc[v] + bias;
    if (relu_flag) val = fmaxf(val, 0.0f);
    op[(size_t)v * DD] = val;
  }
}

// ---------------------------------------------------------------------------
extern "C" void kernel_launch(void* const* d_in, const int* in_sizes, int n_in,
                              void* d_out, int out_size, void* d_ws, size_t ws_size,
                              hipStream_t stream) {
  const float* l_feat = (const float*)d_in[0];
  const float* r_feat = (const float*)d_in[1];
  const int*   rows   = (const int*)d_in[2];
  const int*   cols   = (const int*)d_in[3];
  const float* vals   = (const float*)d_in[4];
  const float* W1a = (const float*)d_in[5];
  const float* b1a = (const float*)d_in[6];
  const float* W2a = (const float*)d_in[7];
  const float* b2a = (const float*)d_in[8];
  const float* W1b = (const float*)d_in[9];
  const float* b1b = (const float*)d_in[10];
  const float* W2b = (const float*)d_in[11];
  const float* b2b = (const float*)d_in[12];

  const size_t M = (size_t)NN * DD;     // elements per feature matrix
  float* B[8];
  for (int i = 0; i < 8; i++) B[i] = (float*)d_ws + (size_t)i * M;

  const int n4 = (int)(M / 4);
  dim3 blk(256);
  dim3 gz((n4 + 255) / 256);            // elementwise / zero grids
  dim3 gs((EE * 32 + 255) / 256);       // 80000 blocks, wave-per-edge
  dim3 gg(NN / 16);                     // 2500 GEMM row-tiles

  // ---------------- layer 1 (needs both out_l and out_r) ----------------
  float *a0 = B[0], *b0 = B[1], *Sl2 = B[2], *Sr2 = B[3];
  float *Kl = B[4], *Kr = B[5], *a1 = B[6], *b1m = B[7];

  zero_f32<<<gz, blk, 0, stream>>>(a0, n4);
  zero_f32<<<gz, blk, 0, stream>>>(b0, n4);
  spmm_atomic<<<gs, blk, 0, stream>>>(rows, cols, vals, r_feat, a0);  // a0 = A  @ fr
  spmm_atomic<<<gs, blk, 0, stream>>>(cols, rows, vals, l_feat, b0);  // b0 = A^T@ fl
  ew_hop0<<<gz, blk, 0, stream>>>(a0, b0, l_feat, r_feat, Sl2, Sr2, Kl, Kr, n4);
  zero_f32<<<gz, blk, 0, stream>>>(a1, n4);
  spmm_atomic<<<gs, blk, 0, stream>>>(rows, cols, vals, Kr, a1);      // a1 = A  @ Kr1
  ew_acc<<<gz, blk, 0, stream>>>(a0, Sl2, a1, Kl, n4);                // Sl1+=a1; Sl2+=a1*Kl1
  zero_f32<<<gz, blk, 0, stream>>>(b1m, n4);
  spmm_atomic<<<gs, blk, 0, stream>>>(cols, rows, vals, Kl, b1m);     // b1 = A^T@ Kl1
  ew_acc<<<gz, blk, 0, stream>>>(b0, Sr2, b1m, Kr, n4);

  float* y1 = Kl;  // Kl/Kr fully consumed above -> reuse for relu'd outputs
  float* z1 = Kr;
  wmma_gemm2<<<gg, blk, 0, stream>>>(a0, Sl2, W1a, W2a, b1a, b2a, y1, 1);
  wmma_gemm2<<<gg, blk, 0, stream>>>(b0, Sr2, W1a, W2a, b1a, b2a, z1, 1);

  // ---------------- layer 2 (only out_l needed) ----------------
  // b-side hop-2 SpMM and Sr accumulation are dead code for out_l -> skipped.
  float *c0 = B[0], *d0 = B[1], *Tl2 = B[2], *Tr2 = B[7];  // Tr2 is a throwaway
  float *Ll = B[3], *Lr = B[6], *c1 = B[7];                // c1 reuses dead Tr2 slot

  zero_f32<<<gz, blk, 0, stream>>>(c0, n4);
  zero_f32<<<gz, blk, 0, stream>>>(d0, n4);
  spmm_atomic<<<gs, blk, 0, stream>>>(rows, cols, vals, z1, c0);      // c0 = A  @ z1
  spmm_atomic<<<gs, blk, 0, stream>>>(cols, rows, vals, y1, d0);      // d0 = A^T@ y1
  ew_hop0<<<gz, blk, 0, stream>>>(c0, d0, y1, z1, Tl2, Tr2, Ll, Lr, n4);
  zero_f32<<<gz, blk, 0, stream>>>(c1, n4);
  spmm_atomic<<<gs, blk, 0, stream>>>(rows, cols, vals, Lr, c1);      // c1 = A  @ Kr1'
  ew_acc<<<gz, blk, 0, stream>>>(c0, Tl2, c1, Ll, n4);
  wmma_gemm2<<<gg, blk, 0, stream>>>(c0, Tl2, W1b, W2b, b1b, b2b, (float*)d_out, 1);
}